// TransformerDecoderKVCache_60902636258021
// MI455X (gfx1250) — compile-verified
//
#include <hip/hip_runtime.h>
#include <stdint.h>

// ---------------------------------------------------------------------------
// Varlen KV-cache append (THD splice) for MI455X / gfx1250.
//
// Pure bandwidth problem (~545 MB total traffic -> ~23 us @ 23.3 TB/s).
// Bulk data streams through the CDNA5 async DMA path:
//   global_load_async_to_lds_b128 (th:NT) -> s_wait_asynccnt ->
//   global_store_async_from_lds_b128 (th:NT)
// so it never occupies VGPRs; each wave keeps 4 x 512B transfers in flight.
// K and V are handled by one launch (blockIdx.y) for occupancy.
// ---------------------------------------------------------------------------

namespace {
constexpr int kH            = 8;
constexpr int kD            = 128;
constexpr int kRowElts      = kH * kD;        // 1024 floats per token-row
constexpr int kRowBytes     = kRowElts * 4;   // 4096 bytes per token-row
constexpr int kRowsPerBlock = 4;              // 4 async loads in flight / wave
constexpr int kThreads      = 256;            // 8 wave32s; 16B per lane per row
}  // namespace

__global__ __launch_bounds__(kThreads)
void kv_splice_async_kernel(const float* __restrict__ past_k,
                            const float* __restrict__ past_v,
                            const float* __restrict__ cur_k,
                            const float* __restrict__ cur_v,
                            const int* __restrict__ past_cu,
                            const int* __restrict__ cur_cu,
                            float* __restrict__ out_k,
                            float* __restrict__ out_v,
                            int B, int T_out)
{
    __shared__ char smem[kRowsPerBlock * kRowBytes];   // 16 KB staging

    const int tid  = threadIdx.x;
    const int row0 = blockIdx.x * kRowsPerBlock;

    // K (y==0) or V (y==1): uniform scalar select.
    const bool  isV  = (blockIdx.y != 0);
    const float* past = isV ? past_v : past_k;
    const float* cur  = isV ? cur_v  : cur_k;
    float*       out  = isV ? out_v  : out_k;

    // LDS byte address for this lane's 16B slot (low 32 bits of the flat
    // address are the LDS offset per the CDNA5 aperture rules).
    const uint32_t lds_lane = (uint32_t)(uintptr_t)(&smem[0]) + (uint32_t)tid * 16u;

    // Find bi = max{i : past_cu[i]+cur_cu[i] <= row0}. out_cu is monotone,
    // so search once compactly, then advance incrementally per row.
    int bi = 0;
#pragma clang loop unroll(disable)
    for (int i = 1; i < B; ++i) {
        if (row0 >= past_cu[i] + cur_cu[i]) bi = i;
        else break;
    }

    // ---- issue async loads: global -> LDS --------------------------------
#pragma unroll
    for (int r = 0; r < kRowsPerBlock; ++r) {
        const int t = row0 + r;                 // output token row (uniform)
        if (t >= T_out) break;                  // uniform branch

        // advance sequence index (rows are consecutive, out_cu monotone)
        while (bi + 1 < B && t >= past_cu[bi + 1] + cur_cu[bi + 1]) ++bi;

        const int p0   = past_cu[bi];
        const int plen = past_cu[bi + 1] - p0;
        const int c0   = cur_cu[bi];
        const int off  = t - (p0 + c0);         // offset inside output seq

        const float* srcRow = (off < plen)
            ? past + (size_t)(p0 + off) * kRowElts
            : cur  + (size_t)(c0 + (off - plen)) * kRowElts;

        const uint64_t gsrc = (uint64_t)(uintptr_t)srcRow + (uint64_t)tid * 16u;
        const uint32_t lds  = lds_lane + (uint32_t)(r * kRowBytes);

        asm volatile("global_load_async_to_lds_b128 %0, %1, off th:TH_LOAD_NT"
                     :
                     : "v"(lds), "v"(gsrc)
                     : "memory");
    }

    // Drain ASYNCcnt: LDS writes from the async loads must land before the
    // async stores read them (async LDS accesses are unordered otherwise).
    asm volatile("s_wait_asynccnt 0x0" ::: "memory");

    // ---- issue async stores: LDS -> global -------------------------------
#pragma unroll
    for (int r = 0; r < kRowsPerBlock; ++r) {
        const int t = row0 + r;
        if (t >= T_out) break;

        const uint64_t gdst =
            (uint64_t)(uintptr_t)(out + (size_t)t * kRowElts) + (uint64_t)tid * 16u;
        const uint32_t lds = lds_lane + (uint32_t)(r * kRowBytes);

        asm volatile("global_store_async_from_lds_b128 %0, %1, off th:TH_STORE_NT"
                     :
                     : "v"(gdst), "v"(lds)
                     : "memory");
    }
    // S_ENDPGM performs an implicit wait-idle, flushing outstanding stores.
}

__global__ void kv_cu_kernel(const int* __restrict__ past_cu,
                             const int* __restrict__ cur_cu,
                             float* __restrict__ out_cu, int n)
{
    const int i = blockIdx.x * blockDim.x + threadIdx.x;
    if (i < n) out_cu[i] = (float)(past_cu[i] + cur_cu[i]);
}

extern "C" void kernel_launch(void* const* d_in, const int* in_sizes, int n_in,
                              void* d_out, int out_size, void* d_ws, size_t ws_size,
                              hipStream_t stream)
{
    (void)n_in; (void)out_size; (void)d_ws; (void)ws_size;

    // setup_inputs() order:
    //   0: past_k [T_past,H,D] f32      1: past_v [T_past,H,D] f32
    //   2: past_cu_seqlens [B+1] i32    3: cur_k  [T_cur,H,D]  f32
    //   4: cur_v  [T_cur,H,D]  f32      5: cur_cu_seqlens [B+1] i32
    const float* past_k  = (const float*)d_in[0];
    const float* past_v  = (const float*)d_in[1];
    const int*   past_cu = (const int*)d_in[2];
    const float* cur_k   = (const float*)d_in[3];
    const float* cur_v   = (const float*)d_in[4];
    const int*   cur_cu  = (const int*)d_in[5];

    const int B      = in_sizes[2] - 1;
    const int T_past = in_sizes[0] / kRowElts;
    const int T_cur  = in_sizes[3] / kRowElts;
    const int T_out  = T_past + T_cur;

    // d_out = new_k | new_v | new_cu (flat, return order)
    float* out_k  = (float*)d_out;
    float* out_v  = out_k + (size_t)T_out * kRowElts;
    float* out_cu = out_v + (size_t)T_out * kRowElts;

    const dim3 grid((T_out + kRowsPerBlock - 1) / kRowsPerBlock, 2);
    kv_splice_async_kernel<<<grid, kThreads, 0, stream>>>(
        past_k, past_v, cur_k, cur_v, past_cu, cur_cu, out_k, out_v, B, T_out);
    kv_cu_kernel<<<1, 64, 0, stream>>>(past_cu, cur_cu, out_cu, B + 1);
}